// SSD_36206574305981
// MI455X (gfx1250) — compile-verified
//
#include <hip/hip_runtime.h>
#include <stdint.h>

#define NBOX   8732
#define NBATCH 128
#define NCLS   21
#define TOPK   200
#define TILE   256
#define PAD    224

// Low 32 bits of a flat pointer to LDS == LDS byte offset (aperture layout:
// LDS flat addr = {shared_base_hi32, lds_offset32}; HW truncates to addr[31:0]).
__device__ __forceinline__ uint32_t lds_lo32(const void* p) {
    return (uint32_t)(uintptr_t)p;
}

__global__ void __launch_bounds__(256) ssd_post_kernel(
        const float* __restrict__ loc,
        const float* __restrict__ conf,
        const float* __restrict__ dbox,
        float* __restrict__ out)
{
    __shared__ uint32_t            s_key[NBOX];          // 34928 B: score bits (0 = invalid)
    __shared__ float               s_tile[TILE * NCLS];  // 21504 B: conf tile / box arrays
    __shared__ unsigned int        s_hist[256];
    __shared__ unsigned long long  s_ck[256];
    __shared__ int s_cnt, s_sel, s_rem, s_valid, s_done;

    // Overlay NMS arrays on s_tile (disjoint lifetimes: s_tile is phase 1 only)
    float* s_x1   = s_tile;
    float* s_y1   = s_tile + PAD;
    float* s_x2   = s_tile + 2 * PAD;
    float* s_y2   = s_tile + 3 * PAD;
    float* s_area = s_tile + 4 * PAD;
    int*   s_keep = (int*)(s_tile + 5 * PAD);

    const int bc  = blockIdx.x;
    const int b   = bc / NCLS;
    const int c   = bc - b * NCLS;
    const int tid = threadIdx.x;
    float* orow = out + (size_t)bc * (TOPK * 5);

    if (c == 0) {  // reference zeroes class 0
        for (int i = tid; i < TOPK * 5; i += 256) orow[i] = 0.0f;
        return;
    }

    if (tid == 0) s_valid = 0;

    // ---- Phase 1: softmax scores for class c via async-to-LDS tiles ----
    const uint32_t tileLds = lds_lo32(s_tile);
    int localValid = 0;

    for (int n0 = 0; n0 < NBOX; n0 += TILE) {
        const int rows = (NBOX - n0 < TILE) ? (NBOX - n0) : TILE;
        const int nv = rows * NCLS / 4;  // rows*21 divisible by 4 for rows=256,28
        const uint64_t gb = (uint64_t)(uintptr_t)(conf + ((size_t)b * NBOX + n0) * NCLS);
        for (int v = tid; v < nv; v += 256) {
            uint32_t la = tileLds + (uint32_t)v * 16u;
            uint32_t go = (uint32_t)v * 16u;
            asm volatile("global_load_async_to_lds_b128 %0, %1, %2"
                         :: "v"(la), "v"(go), "s"(gb) : "memory");
        }
        asm volatile("s_wait_asynccnt 0" ::: "memory");
        __syncthreads();

        if (tid < rows) {
            const float* rp = s_tile + tid * NCLS;  // stride-21 words: conflict free
            float x[NCLS];
            #pragma unroll
            for (int k = 0; k < NCLS; ++k) x[k] = rp[k];
            float m = x[0];
            #pragma unroll
            for (int k = 1; k < NCLS; ++k) m = fmaxf(m, x[k]);
            float s = 0.0f, ec = 0.0f;
            #pragma unroll
            for (int k = 0; k < NCLS; ++k) {
                float ev = expf(x[k] - m);
                s += ev;
                if (k == c) ec = ev;
            }
            float p = ec / s;
            bool valid = p > 0.01f;
            s_key[n0 + tid] = valid ? __float_as_uint(p) : 0u;
            localValid += valid ? 1 : 0;
        }
        __syncthreads();
    }
    atomicAdd(&s_valid, localValid);
    __syncthreads();

    const int kSel = (s_valid < TOPK) ? s_valid : TOPK;
    if (kSel == 0) {
        for (int i = tid; i < TOPK * 5; i += 256) orow[i] = 0.0f;
        return;
    }

    // ---- Phase 2: exact kSel-th largest key64 via radix select w/ early exit ----
    // key64 = (score_bits << 32) | (~n): distinct, descending score then ascending index.
    unsigned long long prefix = 0ull, mask = 0ull;
    int rem = kSel;
    for (int p = 0; p < 8; ++p) {
        const int shift = 56 - 8 * p;
        s_hist[tid] = 0u;
        __syncthreads();
        for (int n = tid; n < NBOX; n += 256) {
            unsigned long long k64 = ((unsigned long long)s_key[n] << 32)
                                   | (unsigned long long)(0xFFFFFFFFu - (uint32_t)n);
            if ((k64 & mask) == prefix)
                atomicAdd(&s_hist[(uint32_t)(k64 >> shift) & 255u], 1u);
        }
        __syncthreads();
        // inclusive suffix-sum (Hillis-Steele)
        for (int st = 1; st < 256; st <<= 1) {
            unsigned int add = (tid + st < 256) ? s_hist[tid + st] : 0u;
            __syncthreads();
            s_hist[tid] += add;
            __syncthreads();
        }
        unsigned int sfx  = s_hist[tid];
        unsigned int sfxn = (tid < 255) ? s_hist[tid + 1] : 0u;
        if (sfx >= (unsigned int)rem && sfxn < (unsigned int)rem) {
            s_sel  = tid;
            s_rem  = rem - (int)sfxn;
            s_done = (sfx == (unsigned int)rem) ? 1 : 0;  // group == need -> T64 = prefix
        }
        __syncthreads();
        prefix |= ((unsigned long long)(unsigned int)s_sel) << shift;
        mask   |= (0xFFull << shift);
        rem = s_rem;
        if (s_done) break;   // uniform
    }
    const unsigned long long T64 = prefix;  // {k64 >= T64} has exactly kSel members

    // ---- Phase 3: gather candidates, bitonic sort descending ----
    s_ck[tid] = 0ull;
    if (tid == 0) s_cnt = 0;
    __syncthreads();
    for (int n = tid; n < NBOX; n += 256) {
        unsigned long long k64 = ((unsigned long long)s_key[n] << 32)
                               | (unsigned long long)(0xFFFFFFFFu - (uint32_t)n);
        if (k64 >= T64) {
            int pos = atomicAdd(&s_cnt, 1);  // exactly kSel hits (keys distinct)
            if (pos < 256) s_ck[pos] = k64;
        }
    }
    __syncthreads();

    for (int k = 2; k <= 256; k <<= 1) {
        for (int j = k >> 1; j > 0; j >>= 1) {
            int ixj = tid ^ j;
            if (ixj > tid) {
                unsigned long long a  = s_ck[tid];
                unsigned long long b2 = s_ck[ixj];
                bool desc = ((tid & k) == 0);
                if ((a < b2) == desc) { s_ck[tid] = b2; s_ck[ixj] = a; }
            }
            __syncthreads();
        }
    }

    // ---- Phase 4: decode candidate boxes (s_tile lifetime over; overlay OK) ----
    float myscore = 0.f, mx1 = 0.f, my1 = 0.f, mx2 = 0.f, my2 = 0.f, marea = 0.f;
    if (tid < PAD) s_keep[tid] = 0;
    if (tid < kSel) {
        unsigned long long k64 = s_ck[tid];
        uint32_t n = 0xFFFFFFFFu - (uint32_t)(k64 & 0xFFFFFFFFull);
        myscore = __uint_as_float((uint32_t)(k64 >> 32));
        float4 l = ((const float4*)loc)[(size_t)b * NBOX + n];
        float4 d = ((const float4*)dbox)[n];
        float cx = d.x + (l.x * 0.1f) * d.z;
        float cy = d.y + (l.y * 0.1f) * d.w;
        float w  = d.z * expf(l.z * 0.2f);
        float h  = d.w * expf(l.w * 0.2f);
        mx1 = cx - w * 0.5f;
        my1 = cy - h * 0.5f;
        mx2 = mx1 + w;
        my2 = my1 + h;
        marea = (mx2 - mx1) * (my2 - my1);
        s_x1[tid] = mx1; s_y1[tid] = my1; s_x2[tid] = mx2; s_y2[tid] = my2;
        s_area[tid] = marea;
        s_keep[tid] = 1;
    }
    __syncthreads();

    // ---- Phase 5: NMS; barrier only on iterations that can write ----
    for (int i = 0; i < kSel - 1; ++i) {
        const bool cur = (s_keep[i] != 0);   // uniform broadcast read
        if (cur) {
            if (tid > i && tid < kSel && s_keep[tid]) {
                float w = fminf(mx2, s_x2[i]) - fmaxf(mx1, s_x1[i]);
                float h = fminf(my2, s_y2[i]) - fmaxf(my1, s_y1[i]);
                w = fmaxf(w, 0.0f); h = fmaxf(h, 0.0f);
                float inter = w * h;
                float iou = inter / (marea - inter + s_area[i]);  // NaN>thr==false, matches ref
                if (iou > 0.45f) s_keep[tid] = 0;
            }
            __syncthreads();                 // uniform (cur identical across block)
        }
    }

    // ---- Phase 6: stable compaction via wave32 ballot + zero fill ----
    const bool mykeep = (tid < kSel) && (s_keep[tid] != 0);
    const uint32_t bal  = (uint32_t)__ballot(mykeep);
    const int wv = tid >> 5, lane = tid & 31;
    if (lane == 0) s_hist[wv] = (unsigned int)__popc(bal);
    __syncthreads();
    int base = 0, total = 0;
    #pragma unroll
    for (int w = 0; w < 8; ++w) {
        int cw = (int)s_hist[w];
        base  += (w < wv) ? cw : 0;
        total += cw;
    }
    if (mykeep) {
        int rank = base + __popc(bal & ((1u << lane) - 1u));
        float* r = orow + rank * 5;
        r[0] = myscore; r[1] = mx1; r[2] = my1; r[3] = mx2; r[4] = my2;
    }
    for (int i = total * 5 + tid; i < TOPK * 5; i += 256) orow[i] = 0.0f;
}

extern "C" void kernel_launch(void* const* d_in, const int* in_sizes, int n_in,
                              void* d_out, int out_size, void* d_ws, size_t ws_size,
                              hipStream_t stream) {
    (void)in_sizes; (void)n_in; (void)out_size; (void)d_ws; (void)ws_size;
    const float* loc  = (const float*)d_in[0];
    const float* conf = (const float*)d_in[1];
    const float* dbox = (const float*)d_in[2];
    ssd_post_kernel<<<NBATCH * NCLS, 256, 0, stream>>>(loc, conf, dbox, (float*)d_out);
}